// HyperButterflyLayer_89670327205916
// MI455X (gfx1250) — compile-verified
//
#include <hip/hip_runtime.h>
#include <math.h>

// ---------------------------------------------------------------------------
// HyperButterflyLayer, MI455X (gfx1250, wave32).
//
// Bandwidth-bound: 512 MB total traffic @ 23.3 TB/s => ~22 us floor; ~0.5
// GFLOP of math. No matmul exists in this op, so WMMA is not applicable; the
// CDNA5-specific data-movement paths used instead are async global->LDS loads
// (+ s_wait_asynccnt) in the row-mixing kernel and global_prefetch_b8 in the
// streaming kernel.
//
// Math (base point 0, c = 1):
//   log map : u   = atanh(||x||)/||x|| * x
//   butterfly touches only rows 0..2045 (param list caps at 1023 pairs):
//     layer 0: overlapping pairs (r, r+1), reads from original u
//     layers 1..9: identical rotation on pairs (2j,2j+1) each layer
//                  => single composed rotation (A,B) = (a + i b)^9
//   exp map : out = tanh(max(||v||,1e-8))/max(||v||,1e-8) * v
// Rows >= 2046: out = s(||x||) * x with one per-row scalar.
// ---------------------------------------------------------------------------

constexpr int kBatch        = 65536;
constexpr int kDim          = 1024;
constexpr int kPairs        = 1023;          // params has 2046 floats
constexpr int kMixedRows    = 2 * kPairs;    // 2046 rows touched by butterfly
constexpr int kTPB          = 256;           // 8 wave32 per block
constexpr int kRowsPerGroup = 4;
constexpr int kTailBlocks   = 4096;

#if defined(__AMDGCN__) && __has_builtin(__builtin_amdgcn_global_load_async_to_lds_b128)
#define USE_ASYNC_LDS 1
#else
#define USE_ASYNC_LDS 0
#endif

// ----------------------------- small helpers ------------------------------

__device__ __forceinline__ float f4_dot(float4 v) {
  return v.x * v.x + v.y * v.y + v.z * v.z + v.w * v.w;
}
__device__ __forceinline__ float4 f4_scale(float4 v, float s) {
  return make_float4(v.x * s, v.y * s, v.z * s, v.w * s);
}
__device__ __forceinline__ float4 f4_axpby(float a, float4 x, float b, float4 y) {
  return make_float4(fmaf(a, x.x, b * y.x), fmaf(a, x.y, b * y.y),
                     fmaf(a, x.z, b * y.z), fmaf(a, x.w, b * y.w));
}

// blockDim.x == kTPB (8 waves of 32); returns the block-wide sum to all lanes
__device__ __forceinline__ float block_reduce_sum(float v, float* red) {
  const int lane = threadIdx.x & 31;
  const int wid  = threadIdx.x >> 5;
#pragma unroll
  for (int off = 16; off > 0; off >>= 1) v += __shfl_down(v, off, 32);
  if (lane == 0) red[wid] = v;
  __syncthreads();
  if (wid == 0) {
    float r = (lane < (kTPB / 32)) ? red[lane] : 0.0f;
#pragma unroll
    for (int off = 4; off > 0; off >>= 1) r += __shfl_down(r, off, 32);
    if (lane == 0) red[0] = r;
  }
  __syncthreads();
  float out = red[0];
  __syncthreads();  // allow red[] reuse by the next reduction
  return out;
}

#if USE_ASYNC_LDS
typedef int v4i_t __attribute__((ext_vector_type(4)));
typedef __attribute__((address_space(1))) v4i_t* gptr_v4i;
typedef __attribute__((address_space(3))) v4i_t* lptr_v4i;

__device__ __forceinline__ void async_copy16(const float* gsrc, float* ldst) {
  // gfx1250 GLOBAL_LOAD_ASYNC_TO_LDS_B128 (tracked with ASYNCcnt)
  __builtin_amdgcn_global_load_async_to_lds_b128(
      (gptr_v4i)gsrc, (lptr_v4i)ldst, /*offset=*/0, /*cpol=*/0);
}
__device__ __forceinline__ void wait_async0() {
#if __has_builtin(__builtin_amdgcn_s_wait_asynccnt)
  __builtin_amdgcn_s_wait_asynccnt(0);
#else
  asm volatile("s_wait_asynccnt 0" ::: "memory");
#endif
}
#endif

// ------------------- kernel 1: compose the 9x rotation --------------------
// Layers 1..9 all apply R_j = [[a,b],[-b,a]], a=params[2j], b=params[2j+1].
// Sequential composition == complex power (a + i b)^9, computed by repeated
// multiplication to match the reference's rounding behavior closely.

__global__ void __launch_bounds__(kTPB) hb_compose_kernel(
    const float* __restrict__ params, float* __restrict__ ab) {
  int j = blockIdx.x * blockDim.x + threadIdx.x;
  if (j >= kPairs) return;
  float a = params[2 * j], b = params[2 * j + 1];
  float A = a, B = b;
#pragma unroll
  for (int i = 1; i < 9; ++i) {  // 9 total applications (layers 1..9)
    float An = A * a - B * b;
    float Bn = A * b + B * a;
    A = An;
    B = Bn;
  }
  ab[2 * j]     = A;
  ab[2 * j + 1] = B;
}

// --------------- kernel 2: butterfly-mixed rows 0..2045 -------------------
// One block per pair j: output rows (2j, 2j+1). Needs u rows 2j, 2j+1 and
// (when j <= 510) 2j+2. Rows are staged into LDS with async loads.

__global__ void __launch_bounds__(kTPB) hb_pair_kernel(
    const float* __restrict__ x, const float* __restrict__ params,
    const float* __restrict__ ab, float* __restrict__ out) {
  __shared__ float red[kTPB / 32];
#if USE_ASYNC_LDS
  __shared__ __align__(16) float rows[3 * kDim];
#endif
  const int  j     = blockIdx.x;       // 0..1022
  const int  tid   = threadIdx.x;
  const int  col   = tid * 4;
  const int  r0    = 2 * j;
  const bool need3 = (j <= 510);       // layer-0 top formula needs u[r+1]

  const float* g0 = x + (size_t)r0 * kDim + col;
  const float* g1 = g0 + kDim;
  const float* g2 = g1 + kDim;

  float4 x0, x1, x2;
#if USE_ASYNC_LDS
  async_copy16(g0, &rows[col]);
  async_copy16(g1, &rows[kDim + col]);
  if (need3) async_copy16(g2, &rows[2 * kDim + col]);
  wait_async0();
  __syncthreads();
  x0 = *reinterpret_cast<const float4*>(&rows[col]);
  x1 = *reinterpret_cast<const float4*>(&rows[kDim + col]);
  x2 = need3 ? *reinterpret_cast<const float4*>(&rows[2 * kDim + col])
             : make_float4(0.f, 0.f, 0.f, 0.f);
#else
  x0 = *reinterpret_cast<const float4*>(g0);
  x1 = *reinterpret_cast<const float4*>(g1);
  x2 = need3 ? *reinterpret_cast<const float4*>(g2)
             : make_float4(0.f, 0.f, 0.f, 0.f);
#endif

  // log map: u = atanh(t)/t * x, per row
  float t0 = sqrtf(block_reduce_sum(f4_dot(x0), red));
  float t1 = sqrtf(block_reduce_sum(f4_dot(x1), red));
  float t2 = sqrtf(block_reduce_sum(f4_dot(x2), red));
  float s0 = (t0 > 1e-12f) ? atanhf(t0) / t0 : 1.0f;
  float s1 = (t1 > 1e-12f) ? atanhf(t1) / t1 : 1.0f;
  float s2 = (t2 > 1e-12f) ? atanhf(t2) / t2 : 1.0f;
  float4 u0 = f4_scale(x0, s0);
  float4 u1 = f4_scale(x1, s1);
  float4 u2 = f4_scale(x2, s2);

  // layer 0 (overlapping pairs, reads from original u)
  float4 w0, w1;
  if (j < 511) {
    float a0 = params[2 * r0],     b0 = params[2 * r0 + 1];
    float a1 = params[2 * r0 + 2], b1 = params[2 * r0 + 3];
    w0 = f4_axpby(a0, u0, b0, u1);       // row r:   a_r*u[r]   + b_r*u[r+1]
    w1 = f4_axpby(a1, u1, b1, u2);
  } else if (j == 511) {
    float a = params[2044], b = params[2045];
    w0 = f4_axpby(a, u0, b, u1);         // row 1022
    w1 = f4_axpby(-b, u0, a, u1);        // row 1023 (special last row)
  } else {
    w0 = u0;                             // rows 1024..2045: layer 0 no-op
    w1 = u1;
  }

  // layers 1..9 composed: rotation by (A,B) on pair (2j, 2j+1)
  float A = ab[2 * j], B = ab[2 * j + 1];
  float4 v0 = f4_axpby(A, w0, B, w1);
  float4 v1 = f4_axpby(-B, w0, A, w1);

  // exp map: out = tanh(max(||v||,1e-8))/max(||v||,1e-8) * v
  float n0 = fmaxf(sqrtf(block_reduce_sum(f4_dot(v0), red)), 1e-8f);
  float n1 = fmaxf(sqrtf(block_reduce_sum(f4_dot(v1), red)), 1e-8f);
  float e0 = tanhf(n0) / n0;
  float e1 = tanhf(n1) / n1;

  *reinterpret_cast<float4*>(out + (size_t)r0 * kDim + col)       = f4_scale(v0, e0);
  *reinterpret_cast<float4*>(out + (size_t)(r0 + 1) * kDim + col) = f4_scale(v1, e1);
}

// --------------- kernel 3: untouched rows 2046..65535 ---------------------
// out = [atanh(t)/t * tanh(vn)/vn] * x, t = ||x||, vn = max(atanh(t), 1e-8).
// Pure streaming: float4 loads/stores, one reduction per row, prefetch of the
// next group while the current one is in flight.

__global__ void __launch_bounds__(kTPB) hb_tail_kernel(
    const float* __restrict__ x, float* __restrict__ out, int numGroups) {
  __shared__ float red[kTPB / 32];
  const int tid = threadIdx.x;

  for (int g = blockIdx.x; g < numGroups; g += gridDim.x) {
    // prefetch the next group's rows (gfx1250 global_prefetch_b8)
    int gn = g + gridDim.x;
    if (gn < numGroups) {
      size_t pr = (size_t)kMixedRows + (size_t)gn * kRowsPerGroup;
#pragma unroll
      for (int i = 0; i < kRowsPerGroup; ++i) {
        if (pr + i < (size_t)kBatch)
          __builtin_prefetch(x + (pr + i) * kDim + tid * 4, 0, 1);
      }
    }

    const size_t baseRow = (size_t)kMixedRows + (size_t)g * kRowsPerGroup;
    float4 v[kRowsPerGroup];
    float  ss[kRowsPerGroup];
#pragma unroll
    for (int i = 0; i < kRowsPerGroup; ++i) {
      size_t row = baseRow + i;
      v[i] = (row < (size_t)kBatch)
                 ? *reinterpret_cast<const float4*>(x + row * kDim + tid * 4)
                 : make_float4(0.f, 0.f, 0.f, 0.f);
      ss[i] = f4_dot(v[i]);
    }
#pragma unroll
    for (int i = 0; i < kRowsPerGroup; ++i) {
      float tot = block_reduce_sum(ss[i], red);   // uniform per block
      size_t row = baseRow + i;
      if (row < (size_t)kBatch) {
        float t  = sqrtf(tot);
        float at = atanhf(t);
        float gl = (t > 1e-12f) ? at / t : 1.0f;  // log-map gain
        float vn = fmaxf(at, 1e-8f);              // ||u|| == atanh(t)
        float s  = gl * (tanhf(vn) / vn);         // * exp-map gain
        *reinterpret_cast<float4*>(out + row * kDim + tid * 4) = f4_scale(v[i], s);
      }
    }
  }
}

// ------------------------------- launcher ---------------------------------

extern "C" void kernel_launch(void* const* d_in, const int* in_sizes, int n_in,
                              void* d_out, int out_size, void* d_ws, size_t ws_size,
                              hipStream_t stream) {
  (void)in_sizes; (void)n_in; (void)out_size; (void)ws_size;
  const float* x      = (const float*)d_in[0];
  const float* params = (const float*)d_in[1];
  float*       out    = (float*)d_out;
  float*       ab     = (float*)d_ws;  // 2046 floats of scratch

  hb_compose_kernel<<<(kPairs + kTPB - 1) / kTPB, kTPB, 0, stream>>>(params, ab);

  // pair kernel depends on ab; same stream => ordered after compose
  hb_pair_kernel<<<kPairs, kTPB, 0, stream>>>(x, params, ab, out);

  const int tailRows = kBatch - kMixedRows;                       // 63490
  const int groups   = (tailRows + kRowsPerGroup - 1) / kRowsPerGroup;
  const int blocks   = groups < kTailBlocks ? groups : kTailBlocks;
  hb_tail_kernel<<<blocks, kTPB, 0, stream>>>(x, out, groups);
}